// HashGrid2D_37383395344981
// MI455X (gfx1250) — compile-verified
//
#include <hip/hip_runtime.h>

// HashGrid2D gather for MI455X (gfx1250).
// Memory-bound: ~336 MB of HBM traffic -> ~14us floor @ 23.3 TB/s.
// Strategy: 32-bit hash math, LDS hash exchange, perfectly coalesced
// 128-bit gathers/stores, global_prefetch_b8 to warm table rows,
// non-temporal hints on streaming traffic to keep the 64 MB table in L2.

#define HASH_MASK ((1u << 19) - 1u)
#define DIM 32
#define PRIME_X 73856093u
#define PRIME_Y 19349663u
#define TILE 256

// Native clang vectors (NOT HIP_vector_type classes) so the
// __builtin_nontemporal_* builtins accept pointers to them.
typedef float v2f __attribute__((ext_vector_type(2)));
typedef float v4f __attribute__((ext_vector_type(4)));

__global__ __launch_bounds__(TILE) void hashgrid_gather_kernel(
    const float* __restrict__ pos,
    const float* __restrict__ table,
    float* __restrict__ out,
    int n)
{
    __shared__ unsigned shash[TILE];

    const int t    = threadIdx.x;
    const int base = blockIdx.x * TILE;
    const int p    = base + t;

    // ---- Phase 1: hash my point, stash in LDS, prefetch my table row ----
    if (p < n) {
        // Streaming read-once: non-temporal 64-bit load (don't pollute L2).
        const v2f* pos2 = (const v2f*)pos;
        v2f xy = __builtin_nontemporal_load(&pos2[p]);

        // floor(x/1.0) -> non-negative int; 32-bit math is exact for the
        // low 19 bits we keep (low bits of product are width-independent).
        unsigned ix = (unsigned)(int)floorf(xy.x);
        unsigned iy = (unsigned)(int)floorf(xy.y);
        unsigned h  = ((ix * PRIME_X) ^ (iy * PRIME_Y)) & HASH_MASK;
        shash[t] = h;

        // Warm both cachelines of the 128B table row toward this WGP
        // while the block barriers (gfx1250 global_prefetch_b8).
        const char* row = (const char*)(table + (size_t)h * DIM);
        __builtin_prefetch(row,      0, 3);
        __builtin_prefetch(row + 64, 0, 3);
    }
    __syncthreads();

    // ---- Phase 2: coalesced 128-bit gather + store ----
    // Tile emits 256 points * 8 float4 = 2048 float4s.
    // Thread t handles f = k*256 + t  => consecutive threads touch
    // consecutive float4s in both table row and output (b128 coalesced).
    const v4f* __restrict__ tab4 = (const v4f*)table;
    v4f* __restrict__ out4       = (v4f*)out;

#pragma unroll
    for (int k = 0; k < 8; ++k) {
        const int f     = k * TILE + t;
        const int lp    = f >> 3;   // local point index 0..255
        const int chunk = f & 7;    // float4 index within the 32-float row
        const int gp    = base + lp;
        if (gp < n) {
            const unsigned h = shash[lp];               // LDS broadcast
            v4f v = tab4[(size_t)h * 8 + chunk];        // L2-resident gather
            // Streaming write-once output: non-temporal 128-bit store.
            __builtin_nontemporal_store(v, &out4[(size_t)gp * 8 + chunk]);
        }
    }
}

extern "C" void kernel_launch(void* const* d_in, const int* in_sizes, int n_in,
                              void* d_out, int out_size, void* d_ws, size_t ws_size,
                              hipStream_t stream) {
    const float* pos   = (const float*)d_in[0];   // N x 2 float32
    const float* table = (const float*)d_in[1];   // 2^19 x 32 float32
    float* out         = (float*)d_out;           // N x 32 float32

    const int n      = in_sizes[0] / 2;
    const int blocks = (n + TILE - 1) / TILE;

    hipLaunchKernelGGL(hashgrid_gather_kernel,
                       dim3(blocks), dim3(TILE), 0, stream,
                       pos, table, out, n);
}